// TransitionUp_50766513438991
// MI455X (gfx1250) — compile-verified
//
#include <hip/hip_runtime.h>

#define DIM1 512
#define DIM2 256
#define DIM_OUT 256
#define B_ 8
#define N_UP 2048
#define N_DOWN 8192
#define N_PG 10240
#define BN_EPS 1e-5f
#define DIST_EPS 1e-6f

typedef __attribute__((ext_vector_type(2))) float v2f;
typedef __attribute__((ext_vector_type(8))) float v8f;

// -----------------------------------------------------------------------------
// Y[M x 256] = X[M x Kdim] @ W[Kdim x 256] + bias
// One wave computes a 64x64 tile: 4 A fragments x 4 B fragments -> 16
// V_WMMA_F32_16X16X4_F32 per K-step (K += 4), so each loaded fragment feeds 4
// WMMAs (16 wmma vs 12 loads per step instead of 4 vs 9 previously).
// Block = 256 threads = 8 waves arranged 2 (M) x 4 (N) -> 128 x 256 block tile.
// Fragment layouts (f32, wave32):
//   A 16x4 : lanes 0-15 hold K=k,k+1 of row M=lane; lanes 16-31 hold K=k+2,k+3
//   B 4x16 : mirrored (rows K across lane halves, col = lane&15)
//   C/D    : VGPR r holds M = r + 8*(lane>=16), N = lane&15
// -----------------------------------------------------------------------------
__global__ __launch_bounds__(256) void gemm_bias_wmma_f32(
    const float* __restrict__ X, const float* __restrict__ W,
    const float* __restrict__ bias, float* __restrict__ Y, int Kdim)
{
  const int lane = threadIdx.x & 31;
  const int wave = threadIdx.x >> 5;       // 0..7
  const int wrow = wave >> 2;              // 0..1
  const int wcol = wave & 3;               // 0..3
  const int row0 = blockIdx.x * 128 + wrow * 64;
  const int col0 = wcol * 64;
  const int m    = lane & 15;              // A row within 16-tile / B col
  const int koff = (lane >> 4) << 1;       // 0 or 2

  const float* __restrict__ xr[4];
#pragma unroll
  for (int i = 0; i < 4; ++i)
    xr[i] = X + (size_t)(row0 + i * 16 + m) * Kdim + koff;

  v8f acc[4][4];
#pragma unroll
  for (int i = 0; i < 4; ++i)
#pragma unroll
    for (int j = 0; j < 4; ++j) acc[i][j] = (v8f){};

  for (int k = 0; k < Kdim; k += 4) {
    v2f a[4];
#pragma unroll
    for (int i = 0; i < 4; ++i) a[i] = *(const v2f*)(xr[i] + k);

    const float* wp = W + (size_t)(k + koff) * DIM_OUT + col0 + m;
    v2f b[4];
#pragma unroll
    for (int j = 0; j < 4; ++j) {
      b[j].x = wp[j * 16];
      b[j].y = wp[DIM_OUT + j * 16];
    }

#pragma unroll
    for (int i = 0; i < 4; ++i)
#pragma unroll
      for (int j = 0; j < 4; ++j)
        acc[i][j] = __builtin_amdgcn_wmma_f32_16x16x4_f32(
            false, a[i], false, b[j], (short)0, acc[i][j], false, false);
  }

  const int rbase = (lane >> 4) << 3;      // 0 or 8
#pragma unroll
  for (int i = 0; i < 4; ++i) {
#pragma unroll
    for (int j = 0; j < 4; ++j) {
      const int col = col0 + j * 16 + m;
      const float bv = bias[col];
#pragma unroll
      for (int r = 0; r < 8; ++r) {
        Y[(size_t)(row0 + i * 16 + rbase + r) * DIM_OUT + col] = acc[i][j][r] + bv;
      }
    }
  }
}

// -----------------------------------------------------------------------------
// BN statistics, deterministic two-stage coalesced reduction.
// Stage 1: each block reduces 256 contiguous rows; thread d owns column d.
// -----------------------------------------------------------------------------
__global__ __launch_bounds__(256) void bn_partial(
    const float* __restrict__ Y, float* __restrict__ ps, float* __restrict__ pq)
{
  const int d = threadIdx.x;
  const size_t r0 = (size_t)blockIdx.x * 256;
  float s = 0.f, q = 0.f;
  for (int r = 0; r < 256; ++r) {
    const float v = Y[(r0 + r) * DIM_OUT + d];
    s += v; q += v * v;
  }
  ps[(size_t)blockIdx.x * DIM_OUT + d] = s;
  pq[(size_t)blockIdx.x * DIM_OUT + d] = q;
}

// Stage 2: fold partials, emit scale = g*rsqrt(var+eps), shift = b - mu*scale.
__global__ __launch_bounds__(256) void bn_finalize(
    const float* __restrict__ ps, const float* __restrict__ pq, int nblk, int rows,
    const float* __restrict__ gamma, const float* __restrict__ beta,
    float* __restrict__ scale, float* __restrict__ shift)
{
  const int d = threadIdx.x;
  float s = 0.f, q = 0.f;
  for (int i = 0; i < nblk; ++i) {
    s += ps[(size_t)i * DIM_OUT + d];
    q += pq[(size_t)i * DIM_OUT + d];
  }
  const float inv = 1.f / (float)rows;
  const float mu  = s * inv;
  const float var = q * inv - mu * mu;
  const float sc  = gamma[d] * rsqrtf(var + BN_EPS);
  scale[d] = sc;
  shift[d] = beta[d] - mu * sc;
}

// -----------------------------------------------------------------------------
// In-place BN + ReLU (y1 -> h1). DIM_OUT is a power of two.
// -----------------------------------------------------------------------------
__global__ __launch_bounds__(256) void bn_relu_apply(
    float* __restrict__ Y, const float* __restrict__ scale,
    const float* __restrict__ shift, int total)
{
  const int i = blockIdx.x * 256 + threadIdx.x;
  if (i < total) {
    const int col = i & (DIM_OUT - 1);
    const float v = Y[i] * scale[col] + shift[col];
    Y[i] = fmaxf(v, 0.f);
  }
}

// -----------------------------------------------------------------------------
// 3-NN search: one thread per down point. pos1 of the batch (24 KB) is staged
// into LDS with CDNA5 GLOBAL_LOAD_ASYNC_TO_LDS_B128 (ASYNCcnt-tracked), then
// each thread keeps a running top-3 min over 2048 points.
// -----------------------------------------------------------------------------
__global__ __launch_bounds__(256) void knn3(
    const float* __restrict__ pos1, const float* __restrict__ pos2,
    int* __restrict__ nidx, float* __restrict__ nw)
{
  __shared__ __align__(16) float sp[N_UP * 3];   // 24576 bytes
  const int b = blockIdx.y;
  const float* p1 = pos1 + (size_t)b * N_UP * 3;

  // 24576 B / 16 B = 1536 b128 transfers; 256 threads x 6 iterations.
  {
    const unsigned lds_base = (unsigned)(size_t)(&sp[0]);   // flat->LDS offset (low 32 bits)
    const unsigned long long gbase = (unsigned long long)(const char*)p1;
#pragma unroll
    for (int it = 0; it < 6; ++it) {
      const unsigned byteoff = ((unsigned)threadIdx.x + (unsigned)it * 256u) * 16u;
      const unsigned lds_addr = lds_base + byteoff;
      const unsigned long long ga = gbase + byteoff;
      asm volatile("global_load_async_to_lds_b128 %0, %1, off"
                   :: "v"(lds_addr), "v"(ga)
                   : "memory");
    }
    asm volatile("s_wait_asynccnt 0x0" ::: "memory");
  }
  __syncthreads();

  const int q = blockIdx.x * 256 + threadIdx.x;            // 0..N_DOWN-1
  const float* dp = pos2 + ((size_t)b * N_PG + N_UP + q) * 3;
  const float qx = dp[0], qy = dp[1], qz = dp[2];

  float bd0 = 3.4e38f, bd1 = 3.4e38f, bd2 = 3.4e38f;
  int bi0 = 0, bi1 = 0, bi2 = 0;
  for (int j = 0; j < N_UP; ++j) {
    const float dx = sp[3 * j + 0] - qx;
    const float dy = sp[3 * j + 1] - qy;
    const float dz = sp[3 * j + 2] - qz;
    const float d = dx * dx + dy * dy + dz * dz;
    if (d < bd0)      { bd2 = bd1; bi2 = bi1; bd1 = bd0; bi1 = bi0; bd0 = d; bi0 = j; }
    else if (d < bd1) { bd2 = bd1; bi2 = bi1; bd1 = d;  bi1 = j; }
    else if (d < bd2) { bd2 = d;  bi2 = j; }
  }
  const float w0 = 1.f / (bd0 + DIST_EPS);
  const float w1 = 1.f / (bd1 + DIST_EPS);
  const float w2 = 1.f / (bd2 + DIST_EPS);
  const float inv = 1.f / (w0 + w1 + w2);
  const size_t o = ((size_t)b * N_DOWN + q) * 3;
  nidx[o + 0] = bi0; nidx[o + 1] = bi1; nidx[o + 2] = bi2;
  nw[o + 0] = w0 * inv; nw[o + 1] = w1 * inv; nw[o + 2] = w2 * inv;
}

// -----------------------------------------------------------------------------
// out[b,n,:] = relu(BN2(y2[b,n,:])) + (n >= N_UP ? sum_k w_k * h1[b, idx_k, :] : 0)
// -----------------------------------------------------------------------------
__global__ __launch_bounds__(256) void fuse_out(
    const float* __restrict__ y2, const float* __restrict__ scale2,
    const float* __restrict__ shift2, const float* __restrict__ h1,
    const int* __restrict__ nidx, const float* __restrict__ nw,
    float* __restrict__ out)
{
  const int n = blockIdx.x;
  const int b = blockIdx.y;
  const int d = threadIdx.x;
  const size_t row = (size_t)b * N_PG + n;

  float v = y2[row * DIM_OUT + d] * scale2[d] + shift2[d];
  v = fmaxf(v, 0.f);

  if (n >= N_UP) {
    const size_t o = ((size_t)b * N_DOWN + (n - N_UP)) * 3;
    const float* hb = h1 + (size_t)b * N_UP * DIM_OUT;
    v += nw[o + 0] * hb[(size_t)nidx[o + 0] * DIM_OUT + d]
       + nw[o + 1] * hb[(size_t)nidx[o + 1] * DIM_OUT + d]
       + nw[o + 2] * hb[(size_t)nidx[o + 2] * DIM_OUT + d];
  }
  out[row * DIM_OUT + d] = v;
}

extern "C" void kernel_launch(void* const* d_in, const int* in_sizes, int n_in,
                              void* d_out, int out_size, void* d_ws, size_t ws_size,
                              hipStream_t stream)
{
  const float* pos1  = (const float*)d_in[0];
  const float* feat1 = (const float*)d_in[1];
  const float* pos2  = (const float*)d_in[2];
  const float* feat2 = (const float*)d_in[3];
  const float* W1    = (const float*)d_in[4];
  const float* b1    = (const float*)d_in[5];
  const float* g1    = (const float*)d_in[6];
  const float* be1   = (const float*)d_in[7];
  const float* W2    = (const float*)d_in[8];
  const float* b2    = (const float*)d_in[9];
  const float* g2    = (const float*)d_in[10];
  const float* be2   = (const float*)d_in[11];
  float* out = (float*)d_out;

  const size_t M1 = (size_t)B_ * N_UP;    // 16384 rows
  const size_t M2 = (size_t)B_ * N_PG;    // 81920 rows
  const int nblk1 = (int)(M1 / 256);      // 64
  const int nblk2 = (int)(M2 / 256);      // 320

  char* ws = (char*)d_ws;
  float* y1     = (float*)ws; ws += M1 * DIM_OUT * sizeof(float);   // 16 MB (becomes h1)
  float* y2     = (float*)ws; ws += M2 * DIM_OUT * sizeof(float);   // 80 MB
  float* scale1 = (float*)ws; ws += DIM_OUT * sizeof(float);
  float* shift1 = (float*)ws; ws += DIM_OUT * sizeof(float);
  float* scale2 = (float*)ws; ws += DIM_OUT * sizeof(float);
  float* shift2 = (float*)ws; ws += DIM_OUT * sizeof(float);
  float* ps1    = (float*)ws; ws += (size_t)nblk1 * DIM_OUT * sizeof(float);
  float* pq1    = (float*)ws; ws += (size_t)nblk1 * DIM_OUT * sizeof(float);
  float* ps2    = (float*)ws; ws += (size_t)nblk2 * DIM_OUT * sizeof(float);
  float* pq2    = (float*)ws; ws += (size_t)nblk2 * DIM_OUT * sizeof(float);
  int*   nidx   = (int*)ws;   ws += (size_t)B_ * N_DOWN * 3 * sizeof(int);
  float* nw     = (float*)ws; ws += (size_t)B_ * N_DOWN * 3 * sizeof(float);

  gemm_bias_wmma_f32<<<dim3((unsigned)(M1 / 128)), dim3(256), 0, stream>>>(feat1, W1, b1, y1, DIM1);
  gemm_bias_wmma_f32<<<dim3((unsigned)(M2 / 128)), dim3(256), 0, stream>>>(feat2, W2, b2, y2, DIM2);

  bn_partial<<<dim3(nblk1), dim3(256), 0, stream>>>(y1, ps1, pq1);
  bn_partial<<<dim3(nblk2), dim3(256), 0, stream>>>(y2, ps2, pq2);
  bn_finalize<<<dim3(1), dim3(256), 0, stream>>>(ps1, pq1, nblk1, (int)M1, g1, be1, scale1, shift1);
  bn_finalize<<<dim3(1), dim3(256), 0, stream>>>(ps2, pq2, nblk2, (int)M2, g2, be2, scale2, shift2);

  bn_relu_apply<<<dim3((unsigned)((M1 * DIM_OUT) / 256)), dim3(256), 0, stream>>>(
      y1, scale1, shift1, (int)(M1 * DIM_OUT));

  knn3<<<dim3(N_DOWN / 256, B_), dim3(256), 0, stream>>>(pos1, pos2, nidx, nw);
  fuse_out<<<dim3(N_PG, B_), dim3(256), 0, stream>>>(y2, scale2, shift2, y1, nidx, nw, out);
}